// LearnedQuantize_71760313582211
// MI455X (gfx1250) — compile-verified
//
#include <hip/hip_runtime.h>
#include <hip/hip_bf16.h>

// ---------------------------------------------------------------------------
// LearnedQuantize for MI455X (gfx1250, wave32).
//   x:     (16, 4096, 64) f32   -> N=65536 rows, D=64
//   embed: (64, 1024)     f32   -> K=1024 codewords
// Outputs (concat): quantize (N*64) | quant_loss (1) | likelihoods (1024)
//
// Score GEMM s[n,k] = 2*x_n . e_k - ||e_k||^2 on V_WMMA_F32_16X16X4_F32.
// ---------------------------------------------------------------------------

typedef __attribute__((ext_vector_type(2))) float v2f;
typedef __attribute__((ext_vector_type(4))) float v4f;
typedef __attribute__((ext_vector_type(8))) float v8f;

#define N_ROWS   65536
#define DIM      64
#define K_EMBED  1024
#define N_CTILES 64      // 1024 / 16
#define N_KCH    16      // 64 / 4

// ---------------------------------------------------------------------------
// Kernel 0: pre-swizzle embed into WMMA B-operand layout + compute ||e_k||^2.
//
// B-matrix (4x16, f32) lane layout assumed symmetric to the documented A
// (16x4) layout: VGPR j, lane l -> B[k = kk*4 + ((l>>4)<<1) + j][n = l&15].
// We pack Bs[t][lane][kk][j] so each lane's 16 k-chunks are 128 contiguous
// bytes -> 8x global_load_b128 per column tile in the main kernel.
// ---------------------------------------------------------------------------
__global__ __launch_bounds__(256) void vq_prep(const float* __restrict__ embed,
                                               float* __restrict__ Bs,
                                               float* __restrict__ normE) {
    int tid = blockIdx.x * 256 + threadIdx.x;   // 65536 threads total
    int j    = tid & 1;
    int kk   = (tid >> 1) & 15;
    int lane = (tid >> 5) & 31;
    int t    = tid >> 10;
    int k = kk * 4 + ((lane >> 4) << 1) + j;    // row of embed (0..63)
    int n = t * 16 + (lane & 15);               // col of embed (0..1023)
    Bs[tid] = embed[k * K_EMBED + n];

    if (tid < K_EMBED) {
        float s = 0.0f;
        #pragma unroll
        for (int d = 0; d < DIM; ++d) {
            float e = embed[d * K_EMBED + tid];
            s += e * e;
        }
        normE[tid] = s;
    }
}

// ---------------------------------------------------------------------------
// Kernel 1: one wave per 16-row stripe. WMMA scores -> LDS -> softmax/argmax.
// ---------------------------------------------------------------------------
__global__ __launch_bounds__(32) void vq_main(const float* __restrict__ x,
                                              const float* __restrict__ embed,
                                              const float* __restrict__ Bs,
                                              const float* __restrict__ normE,
                                              float* __restrict__ out_q,
                                              float* __restrict__ likAccum) {
    __shared__ float lds[16 * K_EMBED];         // 64 KB score stripe

    const int lane    = threadIdx.x;
    const int r15     = lane & 15;
    const int hi      = lane >> 4;              // 0 or 1
    const int rowbase = blockIdx.x * 16;

    // ---- Load A stripe (16 rows x 64 cols), pre-scaled by 2 for the "2*x.e"
    // term. A-matrix 16x4 layout: VGPR j, lane l -> A[m=l&15][kk*4+(hi<<1)+j].
    v2f A[N_KCH];
    const float* xrow = x + (size_t)(rowbase + r15) * DIM + (hi << 1);
    #pragma unroll
    for (int kk = 0; kk < N_KCH; ++kk) {
        v2f a = *(const v2f*)(xrow + kk * 4);
        A[kk] = a * 2.0f;
    }

    // ---- Pass A: 64 column tiles, 16 chained WMMAs each, C seeded with
    // -||e_k||^2 so the accumulator directly holds the softmax logits.
    for (int t = 0; t < N_CTILES; ++t) {
        float cinit = -normE[t * 16 + r15];
        v8f acc;
        #pragma unroll
        for (int v = 0; v < 8; ++v) acc[v] = cinit;

        const v4f* bp = (const v4f*)(Bs + ((size_t)t * 32 + lane) * 32);
        v4f bv[8];
        #pragma unroll
        for (int g = 0; g < 8; ++g) bv[g] = bp[g];

        #pragma unroll
        for (int kk = 0; kk < N_KCH; ++kk) {
            v4f q = bv[kk >> 1];
            v2f b;
            b.x = (kk & 1) ? q.z : q.x;
            b.y = (kk & 1) ? q.w : q.y;
            acc = __builtin_amdgcn_wmma_f32_16x16x4_f32(
                /*neg_a=*/false, A[kk], /*neg_b=*/false, b,
                /*c_mod=*/(short)0, acc, /*reuse_a=*/false, /*reuse_b=*/false);
        }

        // C/D layout: VGPR v, lanes 0-15 -> M=v; lanes 16-31 -> M=v+8; N=l&15.
        #pragma unroll
        for (int v = 0; v < 8; ++v)
            lds[(v + (hi << 3)) * K_EMBED + t * 16 + r15] = acc[v];
    }
    __syncthreads();

    // ---- Pass B: per-row softmax, argmax, likelihood partials.
    float likacc[32];
    #pragma unroll
    for (int i = 0; i < 32; ++i) likacc[i] = 0.0f;

    for (int r = 0; r < 16; ++r) {
        // Sweep: load this row's 1024 scores (lane owns cols g*128+lane*4+ii).
        float sv[32];
        float m = -3.4e38f;
        #pragma unroll
        for (int g = 0; g < 8; ++g) {
            v4f q = *(const v4f*)&lds[r * K_EMBED + g * 128 + lane * 4];
            sv[g * 4 + 0] = q.x; sv[g * 4 + 1] = q.y;
            sv[g * 4 + 2] = q.z; sv[g * 4 + 3] = q.w;
            m = fmaxf(m, fmaxf(fmaxf(q.x, q.y), fmaxf(q.z, q.w)));
        }
        #pragma unroll
        for (int off = 16; off >= 1; off >>= 1)
            m = fmaxf(m, __shfl_xor(m, off, 32));

        // exp + per-lane argmax (cols increase with i -> first-max tie rule).
        float sum = 0.0f, bestv = -3.4e38f;
        int besti = 0;
        #pragma unroll
        for (int i = 0; i < 32; ++i) {
            int col = (i >> 2) * 128 + lane * 4 + (i & 3);
            float s = sv[i];
            if (s > bestv) { bestv = s; besti = col; }
            float e = __expf(s - m);
            sv[i] = e;
            sum += e;
        }
        #pragma unroll
        for (int off = 16; off >= 1; off >>= 1) {
            float ov = __shfl_xor(bestv, off, 32);
            int   oi = __shfl_xor(besti, off, 32);
            if (ov > bestv || (ov == bestv && oi < besti)) { bestv = ov; besti = oi; }
        }
        #pragma unroll
        for (int off = 16; off >= 1; off >>= 1)
            sum += __shfl_xor(sum, off, 32);

        float inv = 1.0f / sum;
        #pragma unroll
        for (int i = 0; i < 32; ++i) likacc[i] += sv[i] * inv;

        // Hard-quantize output: gather winning codeword column.
        int row = rowbase + r;
        int d0 = lane * 2;
        v2f qv;
        qv.x = embed[d0 * K_EMBED + besti];
        qv.y = embed[(d0 + 1) * K_EMBED + besti];
        *(v2f*)&out_q[(size_t)row * DIM + d0] = qv;
    }

    // ---- Flush likelihood partials (each column touched once per wave).
    #pragma unroll
    for (int g = 0; g < 8; ++g)
        #pragma unroll
        for (int ii = 0; ii < 4; ++ii)
            atomicAdd(&likAccum[g * 128 + lane * 4 + ii], likacc[g * 4 + ii]);
}

// ---------------------------------------------------------------------------
// Kernel 2: finalize likelihoods + KL loss.
// ---------------------------------------------------------------------------
__global__ __launch_bounds__(256) void vq_fin(const float* __restrict__ likAccum,
                                              float* __restrict__ out_loss,
                                              float* __restrict__ out_lik) {
    __shared__ float red[256];
    int tid = threadIdx.x;
    const float u = 1.0f / 1024.0f;
    const float logu = -6.9314718055994531f;    // ln(1/1024)
    float part = 0.0f;
    for (int k = tid; k < K_EMBED; k += 256) {
        float lm = likAccum[k] * (1.0f / (float)N_ROWS);
        out_lik[k] = lm;
        part += u * (logu - __logf(lm + 1e-10f));
    }
    red[tid] = part;
    __syncthreads();
    for (int s = 128; s > 0; s >>= 1) {
        if (tid < s) red[tid] += red[tid + s];
        __syncthreads();
    }
    if (tid == 0)
        out_loss[0] = red[0] * (1.0f / (float)K_EMBED) * 0.25f;
}

// ---------------------------------------------------------------------------
extern "C" void kernel_launch(void* const* d_in, const int* in_sizes, int n_in,
                              void* d_out, int out_size, void* d_ws, size_t ws_size,
                              hipStream_t stream) {
    const float* x     = (const float*)d_in[0];   // 16*4096*64
    const float* embed = (const float*)d_in[1];   // 64*1024

    float* out      = (float*)d_out;
    float* out_q    = out;                        // N_ROWS * DIM
    float* out_loss = out + (size_t)N_ROWS * DIM; // 1
    float* out_lik  = out_loss + 1;               // 1024

    float* likAccum = (float*)d_ws;               // 1024 f32
    float* normE    = likAccum + K_EMBED;         // 1024 f32
    float* Bs       = normE + K_EMBED;            // 65536 f32 (swizzled embed)

    hipMemsetAsync(likAccum, 0, K_EMBED * sizeof(float), stream);
    vq_prep<<<256, 256, 0, stream>>>(embed, Bs, normE);
    vq_main<<<N_ROWS / 16, 32, 0, stream>>>(x, embed, Bs, normE, out_q, likAccum);
    vq_fin<<<1, 256, 0, stream>>>(likAccum, out_loss, out_lik);
}